// SimpleGNNRecommender_65712999629519
// MI455X (gfx1250) — compile-verified
//
#include <hip/hip_runtime.h>
#include <hip/hip_bf16.h>

typedef float v2f __attribute__((ext_vector_type(2)));
typedef float v4f __attribute__((ext_vector_type(4)));
typedef float v8f __attribute__((ext_vector_type(8)));

#define NNODES 100000
#define NEDGES 3200000

// ---------------------------------------------------------------- utilities
__global__ void fill_kernel(float* __restrict__ p, float v, int n) {
  int i = blockIdx.x * blockDim.x + threadIdx.x;
  int stride = gridDim.x * blockDim.x;
  for (; i < n; i += stride) p[i] = v;
}

// degree accumulation: deg pre-filled with 1.0 (self loop), +1 per incoming edge
__global__ void edge_deg_kernel(const long long* __restrict__ dst,
                                float* __restrict__ deg, int E) {
  int i = blockIdx.x * blockDim.x + threadIdx.x;
  int stride = gridDim.x * blockDim.x;
  for (; i < E; i += stride) {
    unsafeAtomicAdd(&deg[(int)dst[i]], 1.0f);   // global_atomic_add_f32
  }
}

__global__ void rsqrt_kernel(float* __restrict__ d, int n) {
  int i = blockIdx.x * blockDim.x + threadIdx.x;
  if (i < n) d[i] = rsqrtf(d[i]);               // deg >= 1 always (self loop)
}

// ---------------------------------------------------------------- GEMM (fp32 WMMA)
// HS[m, :] = (X[m, :] @ W) * dis[m]
// One wave per 16-row stripe; accumulate full N (NT 16x16 tiles) reusing A frag.
// V_WMMA_F32_16X16X4_F32 layouts (CDNA5 ISA):
//   A 16x4 : lane = M (l%16); VGPR0/1 = K = 2h / 2h+1  (h = l>>4)
//   B 4x16 : lane = N (l%16); VGPR0/1 = K = 2h / 2h+1
//   C/D    : lane = N (l%16); VGPR v  = row v + 8h
template <int K, int N>
__global__ __launch_bounds__(128) void gemm_scale_kernel(
    const float* __restrict__ X, const float* __restrict__ W,
    const float* __restrict__ dis, float* __restrict__ HS, int M) {
  constexpr int NT = N / 16;
  const int lane = threadIdx.x & 31;
  const int wave = (blockIdx.x * blockDim.x + threadIdx.x) >> 5;
  const int mt = wave;                 // 16-row stripe index
  if (mt * 16 >= M) return;            // wave-uniform guard (EXEC all-1s for WMMA)

  const int h  = lane >> 4;            // half-wave 0/1
  const int ml = lane & 15;
  const int rowA = mt * 16 + ml;

  v8f acc[NT];
#pragma unroll
  for (int t = 0; t < NT; ++t)
#pragma unroll
    for (int j = 0; j < 8; ++j) acc[t][j] = 0.0f;

  const float* __restrict__ Xrow = X + (size_t)rowA * K;

#pragma unroll 4
  for (int k = 0; k < K; k += 4) {
    v2f a = *(const v2f*)(Xrow + k + 2 * h);   // K = k+2h, k+2h+1
#pragma unroll
    for (int t = 0; t < NT; ++t) {
      v2f b;
      b.x = W[(size_t)(k + 2 * h)     * N + t * 16 + ml];
      b.y = W[(size_t)(k + 2 * h + 1) * N + t * 16 + ml];
      acc[t] = __builtin_amdgcn_wmma_f32_16x16x4_f32(
          false, a, false, b, (short)0, acc[t], false, false);
    }
  }

  // D store with per-row dis scaling
#pragma unroll
  for (int v = 0; v < 8; ++v) {
    const int m = mt * 16 + v + 8 * h;
    const float s = dis[m];
#pragma unroll
    for (int t = 0; t < NT; ++t) {
      HS[(size_t)m * N + t * 16 + ml] = acc[t][v] * s;
    }
  }
}

// ---------------------------------------------------------------- edge scatter
// acc[dst, :] += hs[src, :].  One wave per edge; lane owns F/32 contiguous floats.
// hs (51MB) + acc (51MB) are L2-resident (192MB L2) -> runs at L2 rate.
template <int F>
__global__ void edge_agg_kernel(const long long* __restrict__ src,
                                const long long* __restrict__ dst,
                                const float* __restrict__ hs,
                                float* __restrict__ acc, int E) {
  constexpr int V = F / 32;
  const int lane = threadIdx.x & 31;
  long long wave  = ((long long)blockIdx.x * blockDim.x + threadIdx.x) >> 5;
  long long nwave = ((long long)gridDim.x * blockDim.x) >> 5;
  for (long long e = wave; e < E; e += nwave) {
    const int s = (int)src[e];
    const int d = (int)dst[e];
    const float* __restrict__ hrow = hs  + (size_t)s * F + lane * V;
    float* __restrict__       arow = acc + (size_t)d * F + lane * V;
    if constexpr (V == 4) {
      v4f val = *(const v4f*)hrow;                    // global_load_b128 gather
      unsafeAtomicAdd(arow + 0, val.x);
      unsafeAtomicAdd(arow + 1, val.y);
      unsafeAtomicAdd(arow + 2, val.z);
      unsafeAtomicAdd(arow + 3, val.w);
    } else {
      v2f val = *(const v2f*)hrow;
      unsafeAtomicAdd(arow + 0, val.x);
      unsafeAtomicAdd(arow + 1, val.y);
    }
  }
}

// ---------------------------------------------------------------- finalize
// out[i,f] = act( dis[i]*(acc[i,f] + hs[i,f]) + b[f] )
// (dis[i]*hs[i,f] is exactly the self-loop message h[i]*dis[i]^2)
template <int F, bool RELU>
__global__ void finalize_kernel(const float* __restrict__ acc,
                                const float* __restrict__ hs,
                                const float* __restrict__ dis,
                                const float* __restrict__ bias,
                                float* __restrict__ out, int M) {
  long long idx = (long long)blockIdx.x * blockDim.x + threadIdx.x;
  if (idx >= (long long)M * F) return;
  const int i = (int)(idx >> __builtin_ctz(F));
  const int f = (int)(idx & (F - 1));
  float v = dis[i] * (acc[idx] + hs[idx]) + bias[f];
  if (RELU) v = fmaxf(v, 0.0f);
  out[idx] = v;
}

// ---------------------------------------------------------------- launcher
extern "C" void kernel_launch(void* const* d_in, const int* in_sizes, int n_in,
                              void* d_out, int out_size, void* d_ws, size_t ws_size,
                              hipStream_t stream) {
  const float*     x   = (const float*)d_in[0];
  const long long* ei  = (const long long*)d_in[1];   // int64 edge_index [2, E]
  const float*     W1  = (const float*)d_in[2];
  const float*     b1  = (const float*)d_in[3];
  const float*     W2  = (const float*)d_in[4];
  const float*     b2  = (const float*)d_in[5];
  const float*     W3  = (const float*)d_in[6];
  const float*     b3  = (const float*)d_in[7];

  const long long* src = ei;
  const long long* dst = ei + NEDGES;

  float* ws  = (float*)d_ws;
  float* dis = ws;                               // 100000 floats (pad to 102400)
  float* A   = ws + 102400;                      // hs   : N_NODES x 128
  float* B   = A + (size_t)NNODES * 128;         // acc  : N_NODES x 128
  float* C   = B + (size_t)NNODES * 128;         // act  : N_NODES x 128
  float* out = (float*)d_out;

  const int T = 256;
  const int nodeBlocks = (NNODES + T - 1) / T;
  const int feat128    = NNODES * 128;
  const int feat64     = NNODES * 64;
  const int aggBlocks  = (int)(((long long)NEDGES * 32 + T - 1) / T); // 1 wave/edge
  const int gemmBlocks = ((NNODES / 16) * 32 + 127) / 128;            // 6250 waves

  // degrees -> dis = rsqrt(1 + indeg)
  fill_kernel<<<nodeBlocks, T, 0, stream>>>(dis, 1.0f, NNODES);
  edge_deg_kernel<<<4096, T, 0, stream>>>(dst, dis, NEDGES);
  rsqrt_kernel<<<nodeBlocks, T, 0, stream>>>(dis, NNODES);

  // ---- layer 1: 256 -> 128, ReLU
  gemm_scale_kernel<256, 128><<<gemmBlocks, 128, 0, stream>>>(x, W1, dis, A, NNODES);
  fill_kernel<<<(feat128 + T - 1) / T, T, 0, stream>>>(B, 0.0f, feat128);
  edge_agg_kernel<128><<<aggBlocks, T, 0, stream>>>(src, dst, A, B, NEDGES);
  finalize_kernel<128, true><<<(feat128 + T - 1) / T, T, 0, stream>>>(B, A, dis, b1, C, NNODES);

  // ---- layer 2: 128 -> 128, ReLU
  gemm_scale_kernel<128, 128><<<gemmBlocks, 128, 0, stream>>>(C, W2, dis, A, NNODES);
  fill_kernel<<<(feat128 + T - 1) / T, T, 0, stream>>>(B, 0.0f, feat128);
  edge_agg_kernel<128><<<aggBlocks, T, 0, stream>>>(src, dst, A, B, NEDGES);
  finalize_kernel<128, true><<<(feat128 + T - 1) / T, T, 0, stream>>>(B, A, dis, b2, C, NNODES);

  // ---- layer 3: 128 -> 64, no activation
  gemm_scale_kernel<128, 64><<<gemmBlocks, 128, 0, stream>>>(C, W3, dis, A, NNODES);
  fill_kernel<<<(feat64 + T - 1) / T, T, 0, stream>>>(B, 0.0f, feat64);
  edge_agg_kernel<64><<<aggBlocks, T, 0, stream>>>(src, dst, A, B, NEDGES);
  finalize_kernel<64, false><<<(feat64 + T - 1) / T, T, 0, stream>>>(B, A, dis, b3, out, NNODES);
}